// BlockSparseLinear_68410239090948
// MI455X (gfx1250) — compile-verified
//
#include <hip/hip_runtime.h>
#include <stdint.h>

typedef __attribute__((ext_vector_type(2))) float v2f;
typedef __attribute__((ext_vector_type(8))) float v8f;

#define BS 64
#define IN_F 4096
#define OUT_F 4096
#define PAD 68                   // floats/row: bank stride 4 -> conflict-free b64 frag reads
#define XB (BS * PAD)            // 4352 floats: x tile, row-major [r][k]
#define WT (BS * PAD)            // 4352 floats: w tile, TRANSPOSED [n][k]
#define BUFSZ (XB + WT)          // 8704 floats per pipeline buffer

#if defined(__has_builtin) && __has_builtin(__builtin_amdgcn_s_wait_asynccnt)
#define WAIT_ASYNC0() __builtin_amdgcn_s_wait_asynccnt(0)
#else
#define WAIT_ASYNC0() asm volatile("s_wait_asynccnt 0x0" ::: "memory")
#endif

// Async global->LDS copies. LDS operand = wave-relative byte offset = low 32 bits
// of the generic pointer.
__device__ __forceinline__ void async_copy_b128(const float* gsrc, const float* ldst) {
    asm volatile("global_load_async_to_lds_b128 %0, %1, off"
                 :: "v"((uint32_t)(uintptr_t)ldst), "v"(gsrc)
                 : "memory");
}
__device__ __forceinline__ void async_copy_b32(const float* gsrc, const float* ldst) {
    asm volatile("global_load_async_to_lds_b32 %0, %1, off"
                 :: "v"((uint32_t)(uintptr_t)ldst), "v"(gsrc)
                 : "memory");
}

__device__ __forceinline__ int lower_bound(const int* __restrict__ a, int n, int v) {
    int lo = 0, hi = n;
    while (lo < hi) { int m = (lo + hi) >> 1; if (a[m] < v) lo = m + 1; else hi = m; }
    return lo;
}

// Stage one sparse block: x tile row-major via B128; weight block TRANSPOSED via
// per-lane B32 scatter (global side coalesced, async engine does the transpose).
__device__ __forceinline__ void stage_block(const float* __restrict__ x,
                                            const float* __restrict__ w,
                                            int row0, int ib, int k,
                                            float* sx, float* swT, int tid) {
    const float* xsrc = x + (size_t)row0 * IN_F + (size_t)ib * BS;
    const float* wsrc = w + (size_t)k * (BS * BS);
#pragma unroll
    for (int i = 0; i < 4; ++i) {
        const int linear = tid + i * 256;        // 1024 float4 slots
        const int r  = linear >> 4;              // 0..63
        const int c4 = (linear & 15) << 2;       // 0,4,...,60
        async_copy_b128(xsrc + (size_t)r * IN_F + c4, sx + r * PAD + c4);
    }
#pragma unroll
    for (int i = 0; i < 16; ++i) {
        const int linear = tid + i * 256;        // 4096 float slots, coalesced reads
        const int r = linear >> 6;               // weight row (k-dim)
        const int c = linear & 63;               // weight col (n-dim)
        async_copy_b32(wsrc + linear, swT + c * PAD + r);   // transpose scatter
    }
}

__global__ __launch_bounds__(256) void BlockSparseLinear_wmma_async(
    const float* __restrict__ x,
    const float* __restrict__ w,
    const float* __restrict__ bias,
    const int*   __restrict__ out_bi,
    const int*   __restrict__ in_bi,
    float*       __restrict__ out,
    int nnz)
{
    __shared__ float smem[2 * BUFSZ];            // 69,632 B double-buffered

    const int tid  = threadIdx.x;
    const int lane = tid & 31;
    const int wid  = tid >> 5;        // 0..7
    const int hi   = lane >> 4;       // wave half
    const int nl   = lane & 15;
    const int mt   = wid & 3;         // m-tile 0..3
    const int nt0  = (wid >> 2) << 1; // n-tile base 0 or 2

    const int row0 = blockIdx.x * BS;
    const int ob   = blockIdx.y;

    const int mrow  = mt * 16 + nl;   // A-frag row in tile
    const int k0off = hi << 1;        // lane K offset inside K=4 chunk (A and B)
    const int brow0 = (nt0 * 16 + nl) * PAD;       // B-frag rows in transposed w
    const int brow1 = ((nt0 + 1) * 16 + nl) * PAD;

    const int ks = lower_bound(out_bi, nnz, ob);
    const int ke = lower_bound(out_bi, nnz, ob + 1);

    v8f acc0 = {};
    v8f acc1 = {};

    int p = 0;
    if (ks < ke)
        stage_block(x, w, row0, in_bi[ks], ks, smem, smem + XB, tid);

    for (int k = ks; k < ke; ++k) {
        WAIT_ASYNC0();                 // my async writes for block k landed
        __syncthreads();               // everyone's landed; prior compute done
        if (k + 1 < ke) {              // prefetch next block into other buffer
            float* base = smem + (p ^ 1) * BUFSZ;
            stage_block(x, w, row0, in_bi[k + 1], k + 1, base, base + XB, tid);
        }
        const float* sx  = smem + p * BUFSZ;
        const float* swT = sx + XB;
#pragma unroll
        for (int c = 0; c < 16; ++c) {
            const int kb = (c << 2) + k0off;
            const v2f a  = *(const v2f*)(sx  + mrow * PAD + kb);
            const v2f b0 = *(const v2f*)(swT + brow0 + kb);   // contiguous pair
            const v2f b1 = *(const v2f*)(swT + brow1 + kb);
            acc0 = __builtin_amdgcn_wmma_f32_16x16x4_f32(
                false, a, false, b0, (short)0, acc0, false, false);
            acc1 = __builtin_amdgcn_wmma_f32_16x16x4_f32(
                false, a, false, b1, (short)0, acc1, false, false);
        }
        p ^= 1;
    }

    // Epilogue: D layout VGPR v -> M = v + 8*hi, N = lane&15. Stream stores (NT).
    const int col0 = ob * BS + nt0 * 16 + nl;
    const int col1 = col0 + 16;
    const float bv0 = bias[col0];
    const float bv1 = bias[col1];
#pragma unroll
    for (int v = 0; v < 8; ++v) {
        const int r = row0 + mt * 16 + (hi << 3) + v;
        __builtin_nontemporal_store(acc0[v] + bv0, &out[(size_t)r * OUT_F + col0]);
        __builtin_nontemporal_store(acc1[v] + bv1, &out[(size_t)r * OUT_F + col1]);
    }
}

extern "C" void kernel_launch(void* const* d_in, const int* in_sizes, int n_in,
                              void* d_out, int out_size, void* d_ws, size_t ws_size,
                              hipStream_t stream) {
    const float* x      = (const float*)d_in[0];
    const float* weight = (const float*)d_in[1];
    const float* bias   = (const float*)d_in[2];
    const int*   out_bi = (const int*)d_in[3];
    const int*   in_bi  = (const int*)d_in[4];
    float*       out    = (float*)d_out;

    const int nnz  = in_sizes[3];
    const int rows = in_sizes[0] / IN_F;        // 4096
    dim3 grid(rows / BS, OUT_F / BS);           // (64, 64)
    dim3 block(256);
    BlockSparseLinear_wmma_async<<<grid, block, 0, stream>>>(
        x, weight, bias, out_bi, in_bi, out, nnz);
}